// RAM_50560355008534
// MI455X (gfx1250) — compile-verified
//
#include <hip/hip_runtime.h>
#include <math.h>

// Problem constants (match reference)
#define BB 256
#define SS 512
#define EE 768
#define MDD 769        // EE + 1
#define G3E 2304       // 3*EE
#define OUTN 3
#define NW 8           // waves per block (256 threads, wave32)
#define Q4 6           // float4 chunks per lane: 768/(32*4)

typedef __attribute__((ext_vector_type(2))) float v2f;
typedef __attribute__((ext_vector_type(8))) float v8f;

// ---------------------------------------------------------------------------
// Kernel 1: fused locationed-memory + attention scores + online softmax +
// weighted sum. One block per batch element, single pass over memory[b].
// Softmax over s is invariant to the per-b (et.w_et) shift, so alpha (and i)
// is hop-invariant: compute i once.
// ---------------------------------------------------------------------------
__global__ __launch_bounds__(256) void attn_fused_kernel(
    const float* __restrict__ memory, const float* __restrict__ aspect,
    const float* __restrict__ att_w, const float* __restrict__ att_b,
    const int* __restrict__ memory_len, const int* __restrict__ left_len,
    const int* __restrict__ aspect_len, float* __restrict__ i_out)
{
    __shared__ float s_acc[NW][EE];
    __shared__ float s_m[NW], s_l[NW], s_u[NW];
    __shared__ float s_asp;

    const int b    = blockIdx.x;
    const int tid  = threadIdx.x;
    const int lane = tid & 31;
    const int wv   = tid >> 5;

    // ---- aspect . w_asp (w_asp = att_w[MD+E : MD+2E]) ----
    {
        const float* arow = aspect + (size_t)b * EE;
        const float* wasp = att_w + MDD + EE;
        float ap = 0.f;
        for (int e = tid; e < EE; e += 256) ap += arow[e] * wasp[e];
        for (int off = 16; off; off >>= 1) ap += __shfl_xor(ap, off, 32);
        if (lane == 0) s_m[wv] = ap;
        __syncthreads();
        if (tid == 0) {
            float t = 0.f;
            for (int w = 0; w < NW; ++w) t += s_m[w];
            s_asp = t;
        }
        __syncthreads();
    }

    // ---- per-lane slice of w_mem[:768] as float4s ----
    float4 wm4[Q4];
    {
        const float4* w4 = (const float4*)att_w;
        #pragma unroll
        for (int q = 0; q < Q4; ++q) wm4[q] = w4[lane + 32 * q];
    }
    const float wE   = att_w[EE];   // w_mem[768] multiplies u
    const float attb = att_b[0];
    const float asp  = s_asp;
    const int   mlen  = memory_len[b];
    const int   start = left_len[b];
    const int   alen  = aspect_len[b];
    const float inv_mlen = 1.0f / (float)mlen;

    // online softmax state (per wave; acc split across lanes by column)
    float  mCur = -INFINITY, lsum = 0.f, accU = 0.f;
    float4 acc4[Q4];
    #pragma unroll
    for (int q = 0; q < Q4; ++q) acc4[q] = make_float4(0.f, 0.f, 0.f, 0.f);

    const float* base = memory + (size_t)b * SS * EE;
    for (int s = wv; s < SS; s += NW) {
        if (s + NW < SS)
            __builtin_prefetch(base + (size_t)(s + NW) * EE + lane * 4, 0, 1);

        const float4* row4 = (const float4*)(base + (size_t)s * EE);
        float4 r[Q4];
        #pragma unroll
        for (int q = 0; q < Q4; ++q) r[q] = row4[lane + 32 * q];

        float d = 0.f;
        #pragma unroll
        for (int q = 0; q < Q4; ++q)
            d += r[q].x * wm4[q].x + r[q].y * wm4[q].y +
                 r[q].z * wm4[q].z + r[q].w * wm4[q].w;
        for (int off = 16; off; off >>= 1) d += __shfl_xor(d, off, 32);

        // location weights (reference _locationed_memory)
        float uf, lf;
        if (s < start)              { uf = (float)(s - start); lf = (float)(start - s); }
        else if (s < start + alen)  { uf = 0.f; lf = 0.f; }
        else                        { uf = lf = (float)(s - start - alen + 1); }
        float wloc;
        if (s < mlen) { wloc = 1.0f - lf * inv_mlen; }
        else          { wloc = 1.0f; uf = 0.f; }

        const float g = wloc * d + uf * wE + asp + attb;

        const float mNew = fmaxf(mCur, g);
        const float sc   = __expf(mCur - mNew);
        const float p    = __expf(g - mNew);
        lsum = lsum * sc + p;
        accU = accU * sc + p * uf;
        const float pw = p * wloc;
        #pragma unroll
        for (int q = 0; q < Q4; ++q) {
            acc4[q].x = acc4[q].x * sc + pw * r[q].x;
            acc4[q].y = acc4[q].y * sc + pw * r[q].y;
            acc4[q].z = acc4[q].z * sc + pw * r[q].z;
            acc4[q].w = acc4[q].w * sc + pw * r[q].w;
        }
        mCur = mNew;
    }

    // ---- merge 8 wave-local online-softmax states via LDS ----
    {
        float4* dst = (float4*)(&s_acc[wv][0]);
        #pragma unroll
        for (int q = 0; q < Q4; ++q) dst[lane + 32 * q] = acc4[q];
    }
    if (lane == 0) { s_m[wv] = mCur; s_l[wv] = lsum; s_u[wv] = accU; }
    __syncthreads();

    float M = -INFINITY;
    for (int w = 0; w < NW; ++w) M = fmaxf(M, s_m[w]);
    float f[NW], L = 0.f, U = 0.f;
    for (int w = 0; w < NW; ++w) {
        f[w] = __expf(s_m[w] - M);
        L += f[w] * s_l[w];
        U += f[w] * s_u[w];
    }
    const float invL = 1.0f / L;
    float* out = i_out + (size_t)b * MDD;
    for (int c = tid; c < EE; c += 256) {
        float v = 0.f;
        #pragma unroll
        for (int w = 0; w < NW; ++w) v += f[w] * s_acc[w][c];
        out[c] = v * invL;
    }
    if (tid == 0) out[EE] = U * invL;
}

// ---------------------------------------------------------------------------
// GEMM: D[m,n] = sum_k A[m,k] * W[n,k]  via V_WMMA_F32_16X16X4_F32.
// Register-blocked: one 32x32 macro-tile (2x2 WMMA sub-tiles) per wave,
// so each k-step loads 4 fragments (1 KB) and issues 4 WMMAs (8192 FLOP)
// -> 8 FLOP/byte of fragment traffic (2x better than 1 tile/wave).
// ISA layouts (32-bit operands):
//   A 16x4 : lane%16 = M row, VGPR v + 2*(lane/16) = K     -> v2f a
//   B 4x16 : lane%16 = N col, VGPR v + 2*(lane/16) = K     -> v2f b (= W^T)
//   C 16x16: VGPR r, lanes 0-15: M=r, lanes 16-31: M=r+8; N=lane%16
// Tile predicate is wave-uniform -> EXEC all-ones at every WMMA.
// ---------------------------------------------------------------------------
__global__ __launch_bounds__(256) void gemm_tn_wmma_kernel(
    const float* __restrict__ A, int lda,
    const float* __restrict__ W, int ldw,
    float* __restrict__ D, int M, int N, int K)
{
    const int wv   = threadIdx.x >> 5;
    const int lane = threadIdx.x & 31;
    const int tiles_n = N >> 5;                // 32-wide macro tiles
    const int total   = (M >> 5) * tiles_n;
    const int tile    = blockIdx.x * NW + wv;
    if (tile >= total) return;                 // wave-uniform exit

    const int m0   = (tile / tiles_n) << 5;
    const int n0   = (tile % tiles_n) << 5;
    const int half = lane >> 4;
    const int l16  = lane & 15;

    const float* arow0 = A + (size_t)(m0 + l16) * lda + 2 * half;
    const float* arow1 = arow0 + (size_t)16 * lda;
    const float* wrow0 = W + (size_t)(n0 + l16) * ldw + 2 * half;
    const float* wrow1 = wrow0 + (size_t)16 * ldw;

    v8f c00 = {}, c01 = {}, c10 = {}, c11 = {};
    const int kmain = K & ~3;
    for (int k = 0; k < kmain; k += 4) {
        v2f a0, a1, b0, b1;
        a0[0] = arow0[k];  a0[1] = arow0[k + 1];
        a1[0] = arow1[k];  a1[1] = arow1[k + 1];
        b0[0] = wrow0[k];  b0[1] = wrow0[k + 1];
        b1[0] = wrow1[k];  b1[1] = wrow1[k + 1];
        c00 = __builtin_amdgcn_wmma_f32_16x16x4_f32(false, a0, false, b0,
                                                    (short)0, c00, false, false);
        c01 = __builtin_amdgcn_wmma_f32_16x16x4_f32(false, a0, false, b1,
                                                    (short)0, c01, false, false);
        c10 = __builtin_amdgcn_wmma_f32_16x16x4_f32(false, a1, false, b0,
                                                    (short)0, c10, false, false);
        c11 = __builtin_amdgcn_wmma_f32_16x16x4_f32(false, a1, false, b1,
                                                    (short)0, c11, false, false);
    }
    // K remainder (K=769 -> one extra rank-1 update), folded per C layout
    for (int k = kmain; k < K; ++k) {
        const float wk0 = W[(size_t)(n0 + l16) * ldw + k];
        const float wk1 = W[(size_t)(n0 + 16 + l16) * ldw + k];
        #pragma unroll
        for (int r = 0; r < 8; ++r) {
            const float ak0 = A[(size_t)(m0 + r + 8 * half) * lda + k];
            const float ak1 = A[(size_t)(m0 + 16 + r + 8 * half) * lda + k];
            c00[r] += ak0 * wk0;
            c01[r] += ak0 * wk1;
            c10[r] += ak1 * wk0;
            c11[r] += ak1 * wk1;
        }
    }
    #pragma unroll
    for (int r = 0; r < 8; ++r) {
        const size_t mA = (size_t)(m0 + r + 8 * half);
        const size_t mB = mA + 16;
        D[mA * N + n0 + l16]      = c00[r];
        D[mA * N + n0 + 16 + l16] = c01[r];
        D[mB * N + n0 + l16]      = c10[r];
        D[mB * N + n0 + 16 + l16] = c11[r];
    }
}

// ---------------------------------------------------------------------------
// GRU gate update (elementwise): et <- GRUCell(i, et) using precomputed
// gi = i@W_ih^T (hop-invariant) and gh = et@W_hh^T (per hop).
// ---------------------------------------------------------------------------
__global__ __launch_bounds__(256) void gru_gates_kernel(
    const float* __restrict__ gi, const float* __restrict__ gh,
    const float* __restrict__ b_ih, const float* __restrict__ b_hh,
    float* __restrict__ et)
{
    const int idx = blockIdx.x * 256 + threadIdx.x;   // b*E + e
    if (idx >= BB * EE) return;
    const int b = idx / EE, e = idx - b * EE;
    const float* gib = gi + (size_t)b * G3E;
    const float* ghb = gh + (size_t)b * G3E;

    const float rr = gib[e]          + b_ih[e]          + ghb[e]          + b_hh[e];
    const float zz = gib[EE + e]     + b_ih[EE + e]     + ghb[EE + e]     + b_hh[EE + e];
    const float nn_i = gib[2*EE + e] + b_ih[2*EE + e];
    const float nn_h = ghb[2*EE + e] + b_hh[2*EE + e];

    const float r = 1.f / (1.f + __expf(-rr));
    const float z = 1.f / (1.f + __expf(-zz));
    const float n = tanhf(nn_i + r * nn_h);
    const float h = et[idx];
    et[idx] = (1.f - z) * n + z * h;
}

// ---------------------------------------------------------------------------
// logits = et @ dense_w + dense_b   (256x768 @ 768x3)
// ---------------------------------------------------------------------------
__global__ __launch_bounds__(256) void dense_kernel(
    const float* __restrict__ et, const float* __restrict__ dw,
    const float* __restrict__ db, float* __restrict__ out)
{
    __shared__ float sred[OUTN][NW];
    const int b = blockIdx.x, tid = threadIdx.x;
    const int lane = tid & 31, wv = tid >> 5;
    const float* h = et + (size_t)b * EE;
    float p0 = 0.f, p1 = 0.f, p2 = 0.f;
    for (int e = tid; e < EE; e += 256) {
        const float v = h[e];
        p0 += v * dw[e * OUTN + 0];
        p1 += v * dw[e * OUTN + 1];
        p2 += v * dw[e * OUTN + 2];
    }
    for (int off = 16; off; off >>= 1) {
        p0 += __shfl_xor(p0, off, 32);
        p1 += __shfl_xor(p1, off, 32);
        p2 += __shfl_xor(p2, off, 32);
    }
    if (lane == 0) { sred[0][wv] = p0; sred[1][wv] = p1; sred[2][wv] = p2; }
    __syncthreads();
    if (tid < OUTN) {
        float t = db[tid];
        for (int w = 0; w < NW; ++w) t += sred[tid][w];
        out[b * OUTN + tid] = t;
    }
}

__global__ void zero_kernel(float* __restrict__ p, int n)
{
    const int i = blockIdx.x * 256 + threadIdx.x;
    if (i < n) p[i] = 0.f;
}

// ---------------------------------------------------------------------------
extern "C" void kernel_launch(void* const* d_in, const int* in_sizes, int n_in,
                              void* d_out, int out_size, void* d_ws, size_t ws_size,
                              hipStream_t stream)
{
    const float* memory     = (const float*)d_in[0];
    const float* aspect     = (const float*)d_in[1];
    const float* att_w      = (const float*)d_in[2];
    const float* att_b      = (const float*)d_in[3];
    const float* gru_w_ih   = (const float*)d_in[4];
    const float* gru_w_hh   = (const float*)d_in[5];
    const float* gru_b_ih   = (const float*)d_in[6];
    const float* gru_b_hh   = (const float*)d_in[7];
    const float* dense_w    = (const float*)d_in[8];
    const float* dense_b    = (const float*)d_in[9];
    const int*   memory_len = (const int*)d_in[10];
    const int*   left_len   = (const int*)d_in[11];
    const int*   aspect_len = (const int*)d_in[12];
    (void)in_sizes; (void)n_in; (void)ws_size; (void)out_size;

    float* ws    = (float*)d_ws;
    float* i_vec = ws;                       // B*MD
    float* gi    = i_vec + (size_t)BB * MDD; // B*3E
    float* gh    = gi    + (size_t)BB * G3E; // B*3E
    float* et    = gh    + (size_t)BB * G3E; // B*E

    // et = 0
    zero_kernel<<<(BB * EE + 255) / 256, 256, 0, stream>>>(et, BB * EE);

    // one-pass attention -> i (hop-invariant due to softmax shift invariance)
    attn_fused_kernel<<<BB, 256, 0, stream>>>(
        memory, aspect, att_w, att_b, memory_len, left_len, aspect_len, i_vec);

    // gi = i @ W_ih^T  (M=256, N=2304, K=769) — hop-invariant, computed once
    {
        const int blocks = ((BB / 32) * (G3E / 32) + NW - 1) / NW;   // 72
        gemm_tn_wmma_kernel<<<blocks, 256, 0, stream>>>(
            i_vec, MDD, gru_w_ih, MDD, gi, BB, G3E, MDD);
    }

    // 3 GRU hops: gh = et @ W_hh^T then gate update (in-place et)
    for (int hop = 0; hop < 3; ++hop) {
        const int blocks = ((BB / 32) * (G3E / 32) + NW - 1) / NW;   // 72
        gemm_tn_wmma_kernel<<<blocks, 256, 0, stream>>>(
            et, EE, gru_w_hh, EE, gh, BB, G3E, EE);
        gru_gates_kernel<<<(BB * EE + 255) / 256, 256, 0, stream>>>(
            gi, gh, gru_b_ih, gru_b_hh, et);
    }

    // logits
    dense_kernel<<<BB, 256, 0, stream>>>(et, dense_w, dense_b, (float*)d_out);
}